// TDLinear_49598282334337
// MI455X (gfx1250) — compile-verified
//
#include <hip/hip_runtime.h>
#include <hip/hip_bf16.h>
#include <math.h>

#define B_  64
#define S_  512
#define N_  256
#define P_  96
#define D_  64
#define WIN_ 24
#define MID_ 7
#define L_  488      // S - WIN
#define LP_ 496      // L padded to 31 tiles of 16

typedef __attribute__((ext_vector_type(2))) float v2f;
typedef __attribute__((ext_vector_type(8))) float v8f;

__device__ __forceinline__ int imin(int a, int b) { return a < b ? a : b; }

// ---------------------------------------------------------------------------
// 1) Series decomposition: Xt[b,l,n] = mean(X[b,l+1..l+24,n]); Xs = X[b,l+24,n]-Xt
// ---------------------------------------------------------------------------
__global__ __launch_bounds__(256) void decompose_kernel(const float* __restrict__ X,
                                                        float* __restrict__ Xs,
                                                        float* __restrict__ Xt) {
    int n  = threadIdx.x;            // 0..255 (coalesced dim)
    int lc = blockIdx.x;             // 0..15 chunks of 32
    int b  = blockIdx.y;             // 0..63
    int l0 = lc * 32;
    const float* xb = X + (size_t)b * S_ * N_ + n;
    float sum = 0.0f;
    for (int i = 0; i < WIN_; ++i) sum += xb[(size_t)(l0 + 1 + i) * N_];
    for (int j = 0; j < 32; ++j) {
        int l = l0 + j;
        if (l >= L_) break;
        float trend = sum * (1.0f / (float)WIN_);
        float xv = xb[(size_t)(l + WIN_) * N_];
        size_t o = ((size_t)b * L_ + l) * N_ + n;
        Xt[o] = trend;
        Xs[o] = xv - trend;
        if (j < 31 && l + 1 < L_)
            sum += xb[(size_t)(l + WIN_ + 1) * N_] - xb[(size_t)(l + 1) * N_];
    }
}

// ---------------------------------------------------------------------------
// 2) Transpose Xt(B,L,N) -> XtT(N,L,B)  (so trend GEMM B-operand is b-contiguous)
// ---------------------------------------------------------------------------
__global__ __launch_bounds__(256) void transpose_kernel(const float* __restrict__ Xt,
                                                        float* __restrict__ XtT) {
    __shared__ float tile[32][33];
    int l  = blockIdx.z;
    int n0 = blockIdx.x * 32;
    int b0 = blockIdx.y * 32;
    int tx = threadIdx.x, ty = threadIdx.y;
    for (int i = 0; i < 32; i += 8) {
        int b = b0 + ty + i;
        tile[ty + i][tx] = Xt[((size_t)b * L_ + l) * N_ + n0 + tx];
    }
    __syncthreads();
    for (int i = 0; i < 32; i += 8) {
        int n = n0 + ty + i;
        XtT[((size_t)n * L_ + l) * B_ + b0 + tx] = tile[tx][ty + i];
    }
}

// ---------------------------------------------------------------------------
// 3) time2vec embeddings: ET (B,488,64), EN (B,96,64)
// ---------------------------------------------------------------------------
__global__ __launch_bounds__(64) void embed_kernel(const float* __restrict__ T,
                                                   const float* __restrict__ w0,
                                                   const float* __restrict__ b0,
                                                   const float* __restrict__ Wp,
                                                   const float* __restrict__ Bp,
                                                   float* __restrict__ ET,
                                                   float* __restrict__ EN) {
    int d   = threadIdx.x;           // 0..63
    int pos = blockIdx.x;            // 0..583
    int b   = blockIdx.y;
    float t;
    float* dst;
    if (pos < L_) {
        t = T[(size_t)b * S_ + WIN_ + pos];
        dst = ET + ((size_t)b * L_ + pos) * D_ + d;
    } else {
        int p = pos - L_;
        t = T[(size_t)b * S_ + (S_ - 1)] + (float)(p + 1);
        dst = EN + ((size_t)b * P_ + p) * D_ + d;
    }
    float v = (d == 0) ? (t * w0[0] + b0[0]) : sinf(t * Wp[d - 1] + Bp[d - 1]);
    *dst = v;
}

// ---------------------------------------------------------------------------
// 4) Trend: per (n, ptile): out[p,b] = sum_l W_t[n,p,l] * XtT[n,l,b]; += b_t
//    A tile (16xLP_) staged in LDS (coalesced W_t read); 4 waves = 4 b-tiles.
// ---------------------------------------------------------------------------
__global__ __launch_bounds__(128) void trend_kernel(const float* __restrict__ W_t,
                                                    const float* __restrict__ b_t,
                                                    const float* __restrict__ XtT,
                                                    float* __restrict__ out) {
    __shared__ float A[16 * LP_];
    int pt  = blockIdx.x;            // 0..5
    int n   = blockIdx.y;            // 0..255
    int tid = threadIdx.x;
    int p0  = pt * 16;
    const float* wsrc = W_t + ((size_t)n * P_ + p0) * L_;
    for (int idx = tid; idx < 16 * LP_; idx += 128) {
        int r = idx / LP_, c = idx - r * LP_;
        A[idx] = (c < L_) ? wsrc[(size_t)r * L_ + c] : 0.0f;
    }
    __syncthreads();
    int wave = tid >> 5, lane = tid & 31;
    int mrow = lane & 15, khalf = lane >> 4;
    const float* Bb = XtT + (size_t)n * L_ * B_;
    int bb = wave * 16 + mrow;
    v8f acc = {};
    for (int s = 0; s < LP_ / 4; ++s) {
        int k0 = s * 4 + khalf * 2;
        v2f a, bv;
        a.x = A[mrow * LP_ + k0];
        a.y = A[mrow * LP_ + k0 + 1];
        int l0 = imin(k0, L_ - 1), l1 = imin(k0 + 1, L_ - 1);   // pad: A is 0 there
        bv.x = Bb[(size_t)l0 * B_ + bb];
        bv.y = Bb[(size_t)l1 * B_ + bb];
        acc = __builtin_amdgcn_wmma_f32_16x16x4_f32(false, a, false, bv,
                                                    (short)0, acc, false, false);
    }
    for (int v = 0; v < 8; ++v) {
        int p = p0 + v + 8 * khalf;
        out[((size_t)bb * P_ + p) * N_ + n] = acc[v] + b_t[(size_t)p * N_ + n];
    }
}

// ---------------------------------------------------------------------------
// 5) Season attention: per (b, ptile of 16 queries): scores = EN·ETt * 0.125,
//    softmax over 488 keys, out += probs·Xs   (WMMA f32 16x16x4 throughout)
// ---------------------------------------------------------------------------
__global__ __launch_bounds__(128) void season_kernel(const float* __restrict__ ET,
                                                     const float* __restrict__ EN,
                                                     const float* __restrict__ Xs,
                                                     float* __restrict__ out) {
    __shared__ float qs[16 * D_];
    __shared__ float kt[4][16 * D_];       // wave-private K-tile staging
    __shared__ float sc[16 * LP_];         // scores -> probs
    __shared__ float red[16 * 8];
    int pt  = blockIdx.x;                  // 0..5
    int b   = blockIdx.y;                  // 0..63
    int tid = threadIdx.x;
    int wave = tid >> 5, lane = tid & 31;
    int mrow = lane & 15, khalf = lane >> 4;
    int p0 = pt * 16;

    const float* qsrc = EN + ((size_t)b * P_ + p0) * D_;
    for (int idx = tid; idx < 16 * D_; idx += 128) qs[idx] = qsrc[idx];
    __syncthreads();

    const float* et = ET + (size_t)b * L_ * D_;
    // ---- Q·K^T, 31 key tiles, wave-strided (LDS per-wave in-order; no barrier) ----
    for (int nt = wave; nt < LP_ / 16; nt += 4) {
        float* kw = kt[wave];
        int key0 = nt * 16;
        for (int idx = lane; idx < 16 * D_; idx += 32) {
            int kr = idx >> 6, dd = idx & 63;
            int key = key0 + kr;
            kw[idx] = (key < L_) ? et[(size_t)key * D_ + dd] : 0.0f;
        }
        v8f c = {};
        for (int s = 0; s < D_ / 4; ++s) {
            int k0 = s * 4 + khalf * 2;
            v2f a, bv;
            a.x = qs[mrow * D_ + k0];  a.y = qs[mrow * D_ + k0 + 1];
            bv.x = kw[mrow * D_ + k0]; bv.y = kw[mrow * D_ + k0 + 1];
            c = __builtin_amdgcn_wmma_f32_16x16x4_f32(false, a, false, bv,
                                                      (short)0, c, false, false);
        }
        int keyc = key0 + mrow;
        for (int v = 0; v < 8; ++v) {
            int row = v + 8 * khalf;
            sc[row * LP_ + keyc] = (keyc < L_) ? c[v] * 0.125f : -3.0e38f;
        }
    }
    __syncthreads();

    // ---- softmax over LP_ cols (padded cols are -3e38 -> 0) ----
    int r = tid & 15, ch = tid >> 4;       // 8 chunks of 62
    int c0 = ch * (LP_ / 8), c1 = c0 + (LP_ / 8);
    float mx = -3.4e38f;
    for (int cc = c0; cc < c1; ++cc) mx = fmaxf(mx, sc[r * LP_ + cc]);
    red[r * 8 + ch] = mx;
    __syncthreads();
    float rowmax = red[r * 8];
    for (int i = 1; i < 8; ++i) rowmax = fmaxf(rowmax, red[r * 8 + i]);
    __syncthreads();
    float psum = 0.0f;
    for (int cc = c0; cc < c1; ++cc) {
        float e = expf(sc[r * LP_ + cc] - rowmax);
        sc[r * LP_ + cc] = e;
        psum += e;
    }
    red[r * 8 + ch] = psum;
    __syncthreads();
    float rowsum = 0.0f;
    for (int i = 0; i < 8; ++i) rowsum += red[r * 8 + i];
    float inv = 1.0f / rowsum;
    for (int cc = c0; cc < c1; ++cc) sc[r * LP_ + cc] *= inv;
    __syncthreads();

    // ---- probs · Xs, 16 n-tiles, wave-strided; accumulate into d_out ----
    const float* xsb = Xs + (size_t)b * L_ * N_;
    for (int nt = wave; nt < N_ / 16; nt += 4) {
        int n0 = nt * 16 + mrow;
        v8f c = {};
        for (int s = 0; s < LP_ / 4; ++s) {
            int k0 = s * 4 + khalf * 2;
            v2f a, bv;
            a.x = sc[mrow * LP_ + k0];
            a.y = sc[mrow * LP_ + k0 + 1];
            int l0 = imin(k0, L_ - 1), l1 = imin(k0 + 1, L_ - 1);  // probs=0 on pad
            bv.x = xsb[(size_t)l0 * N_ + n0];
            bv.y = xsb[(size_t)l1 * N_ + n0];
            c = __builtin_amdgcn_wmma_f32_16x16x4_f32(false, a, false, bv,
                                                      (short)0, c, false, false);
        }
        for (int v = 0; v < 8; ++v) {
            int p = p0 + v + 8 * khalf;
            size_t o = ((size_t)b * P_ + p) * N_ + n0;
            out[o] += c[v];
        }
    }
}

// ---------------------------------------------------------------------------
// 6) Mid-sequence masked attention (7 queries) -> second output region
// ---------------------------------------------------------------------------
__global__ __launch_bounds__(256) void mid_kernel(const float* __restrict__ ET,
                                                  const float* __restrict__ Xs,
                                                  float* __restrict__ out2) {
    __shared__ float qm[MID_ * D_];
    __shared__ float pr[MID_ * L_];
    __shared__ float red[256];
    int b   = blockIdx.x;
    int tid = threadIdx.x;
    const float* et = ET + (size_t)b * L_ * D_;
    for (int idx = tid; idx < MID_ * D_; idx += 256) {
        int q = idx >> 6, d = idx & 63;
        qm[idx] = et[(size_t)(L_ - MID_ + q) * D_ + d];
    }
    __syncthreads();
    for (int idx = tid; idx < MID_ * L_; idx += 256) {
        int q = idx / L_, k = idx - q * L_;
        float s;
        if (k >= (L_ - MID_ + q)) {
            s = -3.0e38f;                       // mask: only k < t_pos allowed
        } else {
            float acc = 0.0f;
            const float* kv = et + (size_t)k * D_;
            for (int d = 0; d < D_; ++d) acc += qm[q * D_ + d] * kv[d];
            s = acc * 0.125f;
        }
        pr[idx] = s;
    }
    __syncthreads();
    int row = tid >> 5, lane = tid & 31;       // 32 lanes per row, rows 0..6 active
    float mx = -3.4e38f;
    if (row < MID_)
        for (int k = lane; k < L_; k += 32) mx = fmaxf(mx, pr[row * L_ + k]);
    red[tid] = mx;
    __syncthreads();
    float rowmax = -3.4e38f;
    for (int i = 0; i < 32; ++i) rowmax = fmaxf(rowmax, red[(row & 7) * 32 + i]);
    __syncthreads();
    float psum = 0.0f;
    if (row < MID_) {
        for (int k = lane; k < L_; k += 32) {
            float e = expf(pr[row * L_ + k] - rowmax);
            pr[row * L_ + k] = e;
            psum += e;
        }
    }
    red[tid] = psum;
    __syncthreads();
    float rowsum = 0.0f;
    for (int i = 0; i < 32; ++i) rowsum += red[(row & 7) * 32 + i];
    float inv = 1.0f / rowsum;
    if (row < MID_)
        for (int k = lane; k < L_; k += 32) pr[row * L_ + k] *= inv;
    __syncthreads();

    int n = tid;                               // 0..255, coalesced Xs reads
    float acc[MID_];
    for (int q = 0; q < MID_; ++q) acc[q] = 0.0f;
    const float* xsb = Xs + (size_t)b * L_ * N_ + n;
    for (int k = 0; k < L_; ++k) {
        float xv = xsb[(size_t)k * N_];
        for (int q = 0; q < MID_; ++q) acc[q] += pr[q * L_ + k] * xv;
    }
    for (int q = 0; q < MID_; ++q)
        out2[((size_t)b * MID_ + q) * N_ + n] = acc[q];
}

// ---------------------------------------------------------------------------
extern "C" void kernel_launch(void* const* d_in, const int* in_sizes, int n_in,
                              void* d_out, int out_size, void* d_ws, size_t ws_size,
                              hipStream_t stream) {
    const float* X   = (const float*)d_in[0];
    const float* T   = (const float*)d_in[1];
    const float* W_t = (const float*)d_in[2];
    const float* b_t = (const float*)d_in[3];
    const float* w0  = (const float*)d_in[4];
    const float* b0  = (const float*)d_in[5];
    const float* Wp  = (const float*)d_in[6];
    const float* Bp  = (const float*)d_in[7];

    float* ws  = (float*)d_ws;
    float* Xs  = ws;                                   // B*L*N
    float* Xt  = Xs  + (size_t)B_ * L_ * N_;           // B*L*N
    float* XtT = Xt  + (size_t)B_ * L_ * N_;           // N*L*B
    float* ET  = XtT + (size_t)B_ * L_ * N_;           // B*L*D
    float* EN  = ET  + (size_t)B_ * L_ * D_;           // B*P*D

    float* out  = (float*)d_out;                       // (B,P,N)
    float* out2 = out + (size_t)B_ * P_ * N_;          // (B,MID,N)

    decompose_kernel<<<dim3(16, B_), 256, 0, stream>>>(X, Xs, Xt);
    transpose_kernel<<<dim3(N_ / 32, B_ / 32, L_), dim3(32, 8), 0, stream>>>(Xt, XtT);
    embed_kernel<<<dim3(L_ + P_, B_), D_, 0, stream>>>(T, w0, b0, Wp, Bp, ET, EN);
    trend_kernel<<<dim3(P_ / 16, N_), 128, 0, stream>>>(W_t, b_t, XtT, out);
    season_kernel<<<dim3(P_ / 16, B_), 128, 0, stream>>>(ET, EN, Xs, out);
    mid_kernel<<<B_, 256, 0, stream>>>(ET, Xs, out2);
}